// SeebeckGNN_687194767890
// MI455X (gfx1250) — compile-verified
//
#include <hip/hip_runtime.h>

#define THREADS 256

typedef __attribute__((ext_vector_type(2))) float v2f;
typedef __attribute__((ext_vector_type(8))) float v8f;

__device__ __forceinline__ void atomAddF(float* p, float v) {
    __hip_atomic_fetch_add(p, v, __ATOMIC_RELAXED, __HIP_MEMORY_SCOPE_AGENT);
}

// ---------------- degree / norm ----------------
__global__ void k_deg_init(float* deg, int n) {
    int i = blockIdx.x * blockDim.x + threadIdx.x;
    if (i < n) deg[i] = 1.0f;                    // self-loop contributes 1
}

__global__ void k_deg_count(const long long* __restrict__ dst, float* deg, long long E) {
    long long e = (long long)blockIdx.x * blockDim.x + threadIdx.x;
    if (e < E) atomAddF(&deg[(int)dst[e]], 1.0f);
}

__global__ void k_deg_to_dis(float* deg, int n) {
    int i = blockIdx.x * blockDim.x + threadIdx.x;
    if (i < n) deg[i] = rsqrtf(deg[i]);          // deg >= 1 always (self-loop)
}

__global__ void k_edge_norm(const long long* __restrict__ src, const long long* __restrict__ dst,
                            const float* __restrict__ dis, float* __restrict__ norm, long long E) {
    long long e = (long long)blockIdx.x * blockDim.x + threadIdx.x;
    if (e < E) norm[e] = dis[(int)src[e]] * dis[(int)dst[e]];
}

// ---------------- conv1 matmul: h = x @ W1  (K = 2) ----------------
__global__ void k_conv1(const float* __restrict__ x, const float* __restrict__ W1,
                        float* __restrict__ h, long long total) {
    long long t = (long long)blockIdx.x * blockDim.x + threadIdx.x;
    if (t >= total) return;
    long long node = t >> 6;
    int f = (int)(t & 63);
    h[t] = x[node * 2 + 0] * W1[f] + x[node * 2 + 1] * W1[64 + f];
}

// ---- init aggregation buffer with self-loop term + bias: agg = h*dis^2 + b ----
__global__ void k_agg_init(const float* __restrict__ h, const float* __restrict__ dis,
                           const float* __restrict__ b, float* __restrict__ agg, long long total) {
    long long t = (long long)blockIdx.x * blockDim.x + threadIdx.x;
    if (t >= total) return;
    long long node = t >> 6;
    int f = (int)(t & 63);
    float d = dis[node];
    agg[t] = h[t] * d * d + b[f];
}

// ---- edge scatter: agg[dst] += h[src] * norm  (64 lanes = 64 features of one edge) ----
__global__ void k_scatter(const float* __restrict__ h, const long long* __restrict__ src,
                          const long long* __restrict__ dst, const float* __restrict__ norm,
                          float* __restrict__ agg, long long E) {
    long long t = (long long)blockIdx.x * blockDim.x + threadIdx.x;
    long long e = t >> 6;
    if (e >= E) return;
    int f = (int)(t & 63);
    long long s = src[e];
    long long d = dst[e];
    float v = h[s * 64 + f] * norm[e];
    atomAddF(&agg[d * 64 + f], v);
}

// ---------------- conv2 GEMM:  H2 = relu(agg1) @ W2   via fp32 WMMA ----------------
// Per wave: one 16-node tile, K=64, N=64 (4 n-tiles) -> 16 k-steps * 4 = 64 v_wmma.
// fp32 16x16x4 layouts (ISA 7.12.2): A lane<16 holds {K,K+1}, lane>=16 holds {K+2,K+3};
// B mirrors transposed; C/D: VGPR r -> row r (lanes 0-15) / r+8 (lanes 16-31), col = lane&15.
__global__ void __launch_bounds__(256)
k_gemm64_wmma(const float* __restrict__ Ain, const float* __restrict__ W2,
              float* __restrict__ Hout, int nTiles) {
    __shared__ float sW[64 * 64];
    for (int i = threadIdx.x; i < 64 * 64; i += 256) sW[i] = W2[i];
    __syncthreads();

    const int wave = threadIdx.x >> 5;
    const int lane = threadIdx.x & 31;
    const int tile = blockIdx.x * 8 + wave;
    if (tile >= nTiles) return;              // wave-uniform: EXEC all-1s inside

    const int m0   = tile << 4;
    const int half = lane >> 4;              // 0 -> K+0/K+1, 1 -> K+2/K+3
    const int lm   = lane & 15;

    const float* arow = Ain + (size_t)(m0 + lm) * 64;

    v8f acc0 = {}, acc1 = {}, acc2 = {}, acc3 = {};

#pragma unroll
    for (int k = 0; k < 64; k += 4) {
        const int kk = k + (half << 1);
        v2f a;
        a.x = fmaxf(arow[kk + 0], 0.0f);     // fused ReLU of conv1 output
        a.y = fmaxf(arow[kk + 1], 0.0f);
        const float* br0 = &sW[(kk + 0) * 64 + lm];
        const float* br1 = &sW[(kk + 1) * 64 + lm];
        v2f b0; b0.x = br0[0];  b0.y = br1[0];
        v2f b1; b1.x = br0[16]; b1.y = br1[16];
        v2f b2; b2.x = br0[32]; b2.y = br1[32];
        v2f b3; b3.x = br0[48]; b3.y = br1[48];
        acc0 = __builtin_amdgcn_wmma_f32_16x16x4_f32(false, a, false, b0, (short)0, acc0, false, false);
        acc1 = __builtin_amdgcn_wmma_f32_16x16x4_f32(false, a, false, b1, (short)0, acc1, false, false);
        acc2 = __builtin_amdgcn_wmma_f32_16x16x4_f32(false, a, false, b2, (short)0, acc2, false, false);
        acc3 = __builtin_amdgcn_wmma_f32_16x16x4_f32(false, a, false, b3, (short)0, acc3, false, false);
    }

    float* obase = Hout + (size_t)m0 * 64 + lm;
#pragma unroll
    for (int r = 0; r < 8; ++r) {
        float* orow = obase + (size_t)(r + (half << 3)) * 64;
        orow[0]  = acc0[r];
        orow[16] = acc1[r];
        orow[32] = acc2[r];
        orow[48] = acc3[r];
    }
}

// scalar fallback for a non-multiple-of-16 node tail (not taken for N=100000)
__global__ void k_gemm64_tail(const float* __restrict__ Ain, const float* __restrict__ W2,
                              float* __restrict__ Hout, int start, int n) {
    int t = blockIdx.x * blockDim.x + threadIdx.x;
    int node = start + t / 64;
    int f = t % 64;
    if (node >= n) return;
    float s = 0.0f;
    for (int k = 0; k < 64; ++k)
        s += fmaxf(Ain[(size_t)node * 64 + k], 0.0f) * W2[k * 64 + f];
    Hout[(size_t)node * 64 + f] = s;
}

// ---------------- pooling + head ----------------
__global__ void k_zero64(float* p) {
    if (threadIdx.x < 64) p[threadIdx.x] = 0.0f;
}

__global__ void k_pool(const float* __restrict__ agg, float* __restrict__ pooled, long long total) {
    __shared__ float s[THREADS];
    float sum = 0.0f;
    const long long stride = (long long)gridDim.x * blockDim.x;  // multiple of 64
    for (long long i = (long long)blockIdx.x * blockDim.x + threadIdx.x; i < total; i += stride)
        sum += fmaxf(agg[i], 0.0f);                              // fused ReLU of conv2 output
    s[threadIdx.x] = sum;
    __syncthreads();
    if (threadIdx.x < 64) {
        float v = s[threadIdx.x] + s[threadIdx.x + 64] + s[threadIdx.x + 128] + s[threadIdx.x + 192];
        atomAddF(&pooled[threadIdx.x], v);
    }
}

__global__ void k_final(const float* __restrict__ pooled, const float* __restrict__ Wl,
                        const float* __restrict__ bl, float* __restrict__ out, float invN) {
    __shared__ float s[64];
    s[threadIdx.x] = pooled[threadIdx.x] * invN * Wl[threadIdx.x];
    __syncthreads();
    if (threadIdx.x == 0) {
        float t = 0.0f;
        for (int i = 0; i < 64; ++i) t += s[i];
        out[0] = t + bl[0];
    }
}

// ---------------- host side ----------------
extern "C" void kernel_launch(void* const* d_in, const int* in_sizes, int n_in,
                              void* d_out, int out_size, void* d_ws, size_t ws_size,
                              hipStream_t stream) {
    const float*     x   = (const float*)d_in[0];
    const long long* ei  = (const long long*)d_in[1];   // int64 (2, E)
    const float*     W1  = (const float*)d_in[2];
    const float*     b1  = (const float*)d_in[3];
    const float*     W2  = (const float*)d_in[4];
    const float*     b2  = (const float*)d_in[5];
    const float*     Wl  = (const float*)d_in[6];
    const float*     bl  = (const float*)d_in[7];
    float*           out = (float*)d_out;

    const int       n = in_sizes[0] / 2;
    const long long E = in_sizes[1] / 2;
    const long long* src = ei;
    const long long* dst = ei + E;

    auto alignup = [](size_t v) { return (v + 255) & ~(size_t)255; };
    char* p = (char*)d_ws;
    float* dis    = (float*)p; p += alignup((size_t)n * 4);
    float* norm   = (float*)p; p += alignup((size_t)E * 4);
    float* hbuf   = (float*)p; p += alignup((size_t)n * 64 * 4);   // h1, then h2
    float* aggbuf = (float*)p; p += alignup((size_t)n * 64 * 4);   // agg1, then agg2
    float* pooled = (float*)p; p += alignup(64 * 4);

    const long long tot = (long long)n * 64;
    const unsigned bN  = (unsigned)((n + THREADS - 1) / THREADS);
    const unsigned bE  = (unsigned)((E + THREADS - 1) / THREADS);
    const unsigned bT  = (unsigned)((tot + THREADS - 1) / THREADS);
    const unsigned bEF = (unsigned)((E * 64 + THREADS - 1) / THREADS);

    // normalization
    k_deg_init <<<bN, THREADS, 0, stream>>>(dis, n);
    k_deg_count<<<bE, THREADS, 0, stream>>>(dst, dis, E);
    k_deg_to_dis<<<bN, THREADS, 0, stream>>>(dis, n);
    k_edge_norm<<<bE, THREADS, 0, stream>>>(src, dst, dis, norm, E);

    // conv1
    k_conv1   <<<bT, THREADS, 0, stream>>>(x, W1, hbuf, tot);
    k_agg_init<<<bT, THREADS, 0, stream>>>(hbuf, dis, b1, aggbuf, tot);
    k_scatter <<<bEF, THREADS, 0, stream>>>(hbuf, src, dst, norm, aggbuf, E);

    // conv2: h2 = relu(agg1) @ W2 via fp32 WMMA (writes into hbuf)
    const int fullTiles = n / 16;
    if (fullTiles) {
        unsigned bG = (unsigned)((fullTiles + 7) / 8);
        k_gemm64_wmma<<<bG, 256, 0, stream>>>(aggbuf, W2, hbuf, fullTiles);
    }
    const int rem = n % 16;
    if (rem) {
        unsigned bR = (unsigned)((rem * 64 + THREADS - 1) / THREADS);
        k_gemm64_tail<<<bR, THREADS, 0, stream>>>(aggbuf, W2, hbuf, fullTiles * 16, n);
    }
    k_agg_init<<<bT, THREADS, 0, stream>>>(hbuf, dis, b2, aggbuf, tot);
    k_scatter <<<bEF, THREADS, 0, stream>>>(hbuf, src, dst, norm, aggbuf, E);

    // mean pool (with fused ReLU) + linear head
    k_zero64<<<1, 64, 0, stream>>>(pooled);
    k_pool  <<<512, THREADS, 0, stream>>>(aggbuf, pooled, tot);
    k_final <<<1, 64, 0, stream>>>(pooled, Wl, bl, out, 1.0f / (float)n);
}